// MetaDBP_79182017069190
// MI455X (gfx1250) — compile-verified
//
#include <hip/hip_runtime.h>
#include <math.h>

// ---- problem constants (from reference) ----
#define NMODES 2
#define NSTEP  5
#define DTAPS  1001
#define NTAPS  101
#define HID    100
#define NB     4
#define SEQL   500000
#define FS_NOM  1.6e11
#define GAMMA_C 0.0016567
#define DZ_C    400000.0

// ---- kernel tiling constants ----
#define PLANE   502784            // per (b,mode) plane stride: >= SEQL + 2064, mult of 1024
#define HPAD    1040              // padded filter table stride (1031 live entries)
#define KEXT    1016              // Toeplitz K extent = 1001 + 15, multiple of 4
#define NKSTRIDE 104              // padded 101-tap nonlinear kernel stride
#define XSPAN   (2048 + KEXT)     // staged signal span per workgroup = 3064
#define XLDS    3260              // > XSPAN + XSPAN/16

typedef __attribute__((ext_vector_type(2))) float v2f;
typedef __attribute__((ext_vector_type(8))) float v8f;

__device__ __forceinline__ int padidx(int i) { return i + (i >> 4); }

// ============================================================================
// Planarize inputs: (B, SEQ, 2) -> (B, 2, PLANE), zero the tail pad.
// ============================================================================
__global__ __launch_bounds__(256) void prep_x(const float* __restrict__ xre,
                                              const float* __restrict__ xim,
                                              float* __restrict__ xr,
                                              float* __restrict__ xi) {
  int idx = blockIdx.x * 256 + threadIdx.x;
  if (idx >= NB * PLANE) return;
  int b = idx / PLANE, l = idx - b * PLANE;
  float r0, i0, r1, i1;
  if (l < SEQL) {
    const float* pr = xre + ((size_t)b * SEQL + l) * NMODES;
    const float* pi = xim + ((size_t)b * SEQL + l) * NMODES;
    r0 = pr[0]; r1 = pr[1]; i0 = pi[0]; i1 = pi[1];
  } else {
    r0 = r1 = i0 = i1 = 0.f;
  }
  size_t p0 = ((size_t)b * 2) * PLANE + l;
  size_t p1 = p0 + PLANE;
  xr[p0] = r0; xr[p1] = r1;
  xi[p0] = i0; xi[p1] = i1;
}

// ============================================================================
// Per-batch filter prep:
//  - dispersion kernel: direct length-1001 inverse DFT of exp(-i*phase), with
//    fftshift folded in; phase reduced mod 2pi in double once per frequency,
//    DFT twiddle reduced exactly via integer modulus.  Stored zero-padded by
//    15 on each side so the Toeplitz A-tile never needs bounds checks.
//  - nonlinear 2x2x101 kernel via the tiny MLP
//  - coef = gamma*dz*P
// ============================================================================
__global__ __launch_bounds__(256) void prep_filters(
    const float* __restrict__ task, const float* __restrict__ W1,
    const float* __restrict__ b1v, const float* __restrict__ W2,
    const float* __restrict__ b2v, float* __restrict__ hrT,
    float* __restrict__ hiT, float* __restrict__ hnT,
    float* __restrict__ NkT, float* __restrict__ coefT) {
  __shared__ float sPh[DTAPS];
  const int b = blockIdx.x, tid = threadIdx.x;
  const double PI_D = 3.14159265358979323846;
  const double twopi = 2.0 * PI_D;
  const double Fs = (double)task[b * 4 + 2];
  const double c_kms = 299792458.0 / 1000.0;
  const double Fc = 299792458.0 / 1.55e-6;
  const double lamb = c_kms / Fc;
  const double beta2 = -(16.5 * lamb * lamb) / (twopi * 0.5 * 2.0 * c_kms) / 1000.0;
  const double dzn = -(double)DZ_C;

  for (int k = tid; k < DTAPS; k += 256) {
    double fk = ((k <= DTAPS / 2) ? (double)k : (double)(k - DTAPS)) / (double)DTAPS;
    double om = twopi * Fs * fk;
    double ph = 0.5 * beta2 * om * om * dzn;
    sPh[k] = (float)fmod(ph, twopi);
  }
  __syncthreads();

  const float w0 = (float)(twopi / (double)DTAPS);
  for (int j = tid; j < HPAD; j += 256) {
    float hr = 0.f, hi = 0.f;
    if (j >= 15 && j <= 15 + DTAPS - 1) {
      int src = (j - 15 + 501) % DTAPS;  // fftshift for odd N
      int kn = 0;
      float sr = 0.f, si = 0.f;
      for (int k = 0; k < DTAPS; ++k) {
        float ang = w0 * (float)kn - sPh[k];
        float s, c;
        __sincosf(ang, &s, &c);
        sr += c; si += s;
        kn += src; if (kn >= DTAPS) kn -= DTAPS;   // exact mod reduction
      }
      hr = sr * (1.0f / (float)DTAPS);
      hi = si * (1.0f / (float)DTAPS);
    }
    hrT[b * HPAD + j] = hr;
    hiT[b * HPAD + j] = hi;
    hnT[b * HPAD + j] = -hi;
  }

  // nonlinear MLP kernel: Nk[b, o, tap], o = 2*mo + mi
  const float Ts = (float)(FS_NOM / Fs);
  const float Ts2 = Ts * Ts;
  if (tid < NTAPS) {
    float pos = fabsf((float)(tid - NTAPS / 2)) * (1.0f / 200.0f);
    float inp = pos * Ts2;
    float acc0 = 0.f, acc1 = 0.f, acc2 = 0.f, acc3 = 0.f;
    for (int h = 0; h < HID; ++h) {
      float hv = inp * W1[h] + b1v[h];
      hv = hv > 0.f ? hv : 0.f;
      acc0 += hv * W2[h * 4 + 0];
      acc1 += hv * W2[h * 4 + 1];
      acc2 += hv * W2[h * 4 + 2];
      acc3 += hv * W2[h * 4 + 3];
    }
    float e = Ts2 * __expf(-inp);
    NkT[b * 4 * NKSTRIDE + 0 * NKSTRIDE + tid] = (acc0 + b2v[0]) * e;
    NkT[b * 4 * NKSTRIDE + 1 * NKSTRIDE + tid] = (acc1 + b2v[1]) * e;
    NkT[b * 4 * NKSTRIDE + 2 * NKSTRIDE + tid] = (acc2 + b2v[2]) * e;
    NkT[b * 4 * NKSTRIDE + 3 * NKSTRIDE + tid] = (acc3 + b2v[3]) * e;
  }
  if (tid == 0) {
    float Pd = task[b * 4 + 0];
    float P = 0.001f * __powf(10.f, Pd * 0.1f) / (float)NMODES;
    coefT[b] = (float)(GAMMA_C * DZ_C) * P;
  }
}

// ============================================================================
// Dispersion step: complex 1001-tap correlation via V_WMMA_F32_16X16X4_F32.
// Workgroup = 256 threads (8 waves), handles one batch, BOTH modes, 2048
// consecutive outputs (each wave owns a 16x16 tile = 256 outputs).
//   D[m,n] = y[base + 16n + m] = sum_t A[m,t] B[t,n]
//   A[m,t] = hpad[15 + t - m]      (Toeplitz filter slab, shared A tiles)
//   B[t,n] = x[base + 16n + t]     (LDS-staged, 17/16 padded vs bank conflicts)
//   ar = xr*hr + xi*(-hi),  ai = xr*hi + xi*hr   -> 8 wmmas / K-step of 4
// ============================================================================
__global__ __launch_bounds__(256) void disp_wmma(
    const float* __restrict__ xr, const float* __restrict__ xi,
    float* __restrict__ ar, float* __restrict__ ai,
    const float* __restrict__ hrT, const float* __restrict__ hiT,
    const float* __restrict__ hnT) {
  __shared__ float sHr[HPAD], sHi[HPAD], sHn[HPAD];
  __shared__ float sXr0[XLDS], sXi0[XLDS], sXr1[XLDS], sXi1[XLDS];
  const int tid = threadIdx.x;
  const int b = blockIdx.y;
  const size_t p0 = ((size_t)b * 2) * PLANE;
  const size_t p1 = p0 + PLANE;
  const int wgBase = blockIdx.x * 2048;

  for (int i = tid; i < HPAD; i += 256) {
    sHr[i] = hrT[b * HPAD + i];
    sHi[i] = hiT[b * HPAD + i];
    sHn[i] = hnT[b * HPAD + i];
  }
  for (int i = tid; i < XSPAN; i += 256) {
    int fi = padidx(i);
    sXr0[fi] = xr[p0 + wgBase + i];
    sXi0[fi] = xi[p0 + wgBase + i];
    sXr1[fi] = xr[p1 + wgBase + i];
    sXi1[fi] = xi[p1 + wgBase + i];
  }
  __syncthreads();

  const int w = tid >> 5, l = tid & 31;
  const int mn = l & 15;          // A row m / B,D column n
  const int half = l >> 4;        // K pair select per ISA 16x4/4x16 layout
  const int aOff = 15 + 2 * half - mn;
  const int bOff = w * 256 + 16 * mn + 2 * half;

  v8f accR0 = {}, accI0 = {}, accR1 = {}, accI1 = {};
  for (int kb = 0; kb < KEXT; kb += 4) {
    int a = aOff + kb;
    v2f Ahr, Ahi, Ahn;
    Ahr.x = sHr[a]; Ahr.y = sHr[a + 1];
    Ahi.x = sHi[a]; Ahi.y = sHi[a + 1];
    Ahn.x = sHn[a]; Ahn.y = sHn[a + 1];
    int ib = bOff + kb;           // even, (ib&15)!=15 -> padded pair contiguous
    int fb = padidx(ib);
    v2f Br0, Bi0, Br1, Bi1;
    Br0.x = sXr0[fb]; Br0.y = sXr0[fb + 1];
    Bi0.x = sXi0[fb]; Bi0.y = sXi0[fb + 1];
    Br1.x = sXr1[fb]; Br1.y = sXr1[fb + 1];
    Bi1.x = sXi1[fb]; Bi1.y = sXi1[fb + 1];
    accR0 = __builtin_amdgcn_wmma_f32_16x16x4_f32(false, Ahr, false, Br0, (short)0, accR0, false, false);
    accR0 = __builtin_amdgcn_wmma_f32_16x16x4_f32(false, Ahn, false, Bi0, (short)0, accR0, false, false);
    accI0 = __builtin_amdgcn_wmma_f32_16x16x4_f32(false, Ahi, false, Br0, (short)0, accI0, false, false);
    accI0 = __builtin_amdgcn_wmma_f32_16x16x4_f32(false, Ahr, false, Bi0, (short)0, accI0, false, false);
    accR1 = __builtin_amdgcn_wmma_f32_16x16x4_f32(false, Ahr, false, Br1, (short)0, accR1, false, false);
    accR1 = __builtin_amdgcn_wmma_f32_16x16x4_f32(false, Ahn, false, Bi1, (short)0, accR1, false, false);
    accI1 = __builtin_amdgcn_wmma_f32_16x16x4_f32(false, Ahi, false, Br1, (short)0, accI1, false, false);
    accI1 = __builtin_amdgcn_wmma_f32_16x16x4_f32(false, Ahr, false, Bi1, (short)0, accI1, false, false);
  }

  // D layout: VGPR j -> M = j + 8*half, N = mn  ->  y[base + 16*mn + 8*half + j]
  const size_t outBase = (size_t)wgBase + w * 256 + 16 * mn + 8 * half;
#pragma unroll
  for (int j = 0; j < 8; ++j) {
    ar[p0 + outBase + j] = accR0[j];
    ai[p0 + outBase + j] = accI0[j];
    ar[p1 + outBase + j] = accR1[j];
    ai[p1 + outBase + j] = accI1[j];
  }
}

// ============================================================================
// Nonlinear step: p = ar^2+ai^2, phi = 2x2 101-tap conv, theta = phi*coef,
// rotate trimmed (ar,ai) by theta -> new (xr,xi) in place.
// ============================================================================
__global__ __launch_bounds__(256) void nonlin_rotate(
    const float* __restrict__ ar, const float* __restrict__ ai,
    float* __restrict__ xr, float* __restrict__ xi,
    const float* __restrict__ NkT, const float* __restrict__ coefT, int Ln) {
  __shared__ float sNk[4 * NKSTRIDE];
  __shared__ float sP0[360], sP1[360];
  const int tid = threadIdx.x;
  const int b = blockIdx.y;
  const size_t p0 = ((size_t)b * 2) * PLANE;
  const size_t p1 = p0 + PLANE;
  const int t0 = blockIdx.x * 256;

  for (int i = tid; i < 4 * NKSTRIDE; i += 256) sNk[i] = NkT[b * 4 * NKSTRIDE + i];
  {
    int j = t0 + tid;
    float a0r = ar[p0 + j], a0i = ai[p0 + j];
    float a1r = ar[p1 + j], a1i = ai[p1 + j];
    sP0[tid] = a0r * a0r + a0i * a0i;
    sP1[tid] = a1r * a1r + a1i * a1i;
    if (tid < 100) {
      int j2 = j + 256;
      float b0r = ar[p0 + j2], b0i = ai[p0 + j2];
      float b1r = ar[p1 + j2], b1i = ai[p1 + j2];
      sP0[tid + 256] = b0r * b0r + b0i * b0i;
      sP1[tid + 256] = b1r * b1r + b1i * b1i;
    }
  }
  __syncthreads();

  float phi0 = 0.f, phi1 = 0.f;
  for (int k = 0; k < NTAPS; ++k) {
    float pa = sP0[tid + k], pb = sP1[tid + k];
    phi0 += pa * sNk[k] + pb * sNk[NKSTRIDE + k];
    phi1 += pa * sNk[2 * NKSTRIDE + k] + pb * sNk[3 * NKSTRIDE + k];
  }
  float coef = coefT[b];
  float s0, c0, s1, c1;
  __sincosf(phi0 * coef, &s0, &c0);
  __sincosf(phi1 * coef, &s1, &c1);
  int n = t0 + tid;
  float a0r = ar[p0 + n + 50], a0i = ai[p0 + n + 50];
  float a1r = ar[p1 + n + 50], a1i = ai[p1 + n + 50];
  if (n < Ln) {
    xr[p0 + n] = a0r * c0 - a0i * s0;
    xi[p0 + n] = a0r * s0 + a0i * c0;
    xr[p1 + n] = a1r * c1 - a1i * s1;
    xi[p1 + n] = a1r * s1 + a1i * c1;
  }
}

// ============================================================================
// Write (B, Lf, NMODES, 2) output from planar buffers as float4 per (b,l).
// ============================================================================
__global__ __launch_bounds__(256) void writeout(const float* __restrict__ xr,
                                                const float* __restrict__ xi,
                                                float4* __restrict__ out, int Lf) {
  int idx = blockIdx.x * 256 + threadIdx.x;
  if (idx >= NB * Lf) return;
  int b = idx / Lf, l = idx - b * Lf;
  size_t p0 = ((size_t)b * 2) * PLANE + l;
  size_t p1 = p0 + PLANE;
  out[idx] = make_float4(xr[p0], xi[p0], xr[p1], xi[p1]);
}

// ============================================================================
// Host launch.  Workspace layout (floats):
//   xR, xI, aR, aI : 4 planes of NB*2*PLANE  (~64.4 MB total)
//   hrT/hiT/hnT    : NB*1040 each, NkT: NB*416, coefT: NB
// ============================================================================
extern "C" void kernel_launch(void* const* d_in, const int* in_sizes, int n_in,
                              void* d_out, int out_size, void* d_ws, size_t ws_size,
                              hipStream_t stream) {
  (void)in_sizes; (void)n_in; (void)out_size; (void)ws_size;
  const float* x_real = (const float*)d_in[0];
  const float* x_imag = (const float*)d_in[1];
  const float* task   = (const float*)d_in[2];
  const float* W1     = (const float*)d_in[3];
  const float* b1v    = (const float*)d_in[4];
  const float* W2     = (const float*)d_in[5];
  const float* b2v    = (const float*)d_in[6];

  float* W = (float*)d_ws;
  const size_t BPN = (size_t)NB * 2 * PLANE;
  float* xR    = W;
  float* xI    = xR + BPN;
  float* aR    = xI + BPN;
  float* aI    = aR + BPN;
  float* hrT   = aI + BPN;
  float* hiT   = hrT + (size_t)NB * HPAD;
  float* hnT   = hiT + (size_t)NB * HPAD;
  float* NkT   = hnT + (size_t)NB * HPAD;
  float* coefT = NkT + (size_t)NB * 4 * NKSTRIDE;

  prep_x<<<(NB * PLANE + 255) / 256, 256, 0, stream>>>(x_real, x_imag, xR, xI);
  prep_filters<<<NB, 256, 0, stream>>>(task, W1, b1v, W2, b2v, hrT, hiT, hnT, NkT, coefT);

  int L = SEQL;
  for (int s = 0; s < NSTEP; ++s) {
    int Lc = L - (DTAPS - 1);
    int Ln = Lc - (NTAPS - 1);
    dim3 gd((unsigned)((Lc + 2047) / 2048), NB);
    disp_wmma<<<gd, 256, 0, stream>>>(xR, xI, aR, aI, hrT, hiT, hnT);
    dim3 gn((unsigned)((Ln + 255) / 256), NB);
    nonlin_rotate<<<gn, 256, 0, stream>>>(aR, aI, xR, xI, NkT, coefT, Ln);
    L = Ln;
  }
  writeout<<<(NB * L + 255) / 256, 256, 0, stream>>>(xR, xI, (float4*)d_out, L);
}